// GraphUnet_12352325943418
// MI455X (gfx1250) — compile-verified
//
#include <hip/hip_runtime.h>
#include <hip/hip_bf16.h>
#include <stdint.h>

// ---------------------------------------------------------------------------
// Graph U-Net forward for MI455X (gfx1250, wave32).
// All GEMMs (GIN feature matmuls, mask^T aggregation, Laplacian powers, 2-hop
// booleanization) run through V_WMMA_F32_16X16X4_F32 with 32x32 per-wave
// tiles (2x2 accumulators, shared A/B fragments -> 4 WMMA per 6 loads) and a
// software-pipelined K-loop (next iteration's fragments load while current
// WMMAs execute). The data-dependent greedy eigh selection is an on-device
// cyclic Jacobi eigensolver, one single-workgroup launch per greedy step.
// ---------------------------------------------------------------------------

typedef __attribute__((ext_vector_type(2))) float v2f;
typedef __attribute__((ext_vector_type(8))) float v8f;

#define EPI_NONE 0
#define EPI_ADDX 1           // C = acc + X
#define EPI_BIAS_RELU 2      // C = relu(acc + bias[col])
#define EPI_BIAS_RELU_ADD 3  // C = relu(acc + bias[col]) + X
#define EPI_BOOL 4           // C = acc > 0 ? 1 : 0

// Block = 8 waves; wave computes a 32x32 tile (grid.y = M/32, block covers
// 256 columns). All M and N in this problem are multiples of 32, so the only
// control flow is a wave-uniform early exit (EXEC all-ones for WMMA).
// A-frag per ISA layout: lanes 0-15 hold (K=k,k+1), lanes 16-31 (K=k+2,k+3).
template <int EPI, bool TRANSA>
__global__ __launch_bounds__(256) void wmma_gemm_f32(
    const float* __restrict__ A, int lda,
    const float* __restrict__ B, int ldb,
    const float* __restrict__ bias,
    const float* __restrict__ X, int ldx,
    float* __restrict__ C, int ldc,
    int M, int N, int K)
{
  const int wave = threadIdx.x >> 5;
  const int lane = threadIdx.x & 31;
  const int m0 = blockIdx.y << 5;                  // 32 rows per wave tile
  const int n0 = (blockIdx.x << 8) + (wave << 5);  // 32 cols per wave tile
  if (n0 >= N) return;  // wave-uniform

  const int mr = lane & 15;
  const int kh = (lane >> 4) << 1;  // 0 or 2

  v8f acc00 = {0.f,0.f,0.f,0.f,0.f,0.f,0.f,0.f};
  v8f acc01 = acc00, acc10 = acc00, acc11 = acc00;

  const float* Ap = TRANSA ? (A + (size_t)kh * lda + (m0 + mr))
                           : (A + (size_t)(m0 + mr) * lda + kh);
  const float* Bp = B + (size_t)kh * ldb + (n0 + mr);
  const size_t astep = TRANSA ? (size_t)4 * lda : (size_t)4;
  const size_t bstep = (size_t)4 * ldb;

  auto loadA = [&](const float* p, v2f& x0, v2f& x1) {
    if (TRANSA) {
      x0.x = p[0];
      x0.y = p[lda];
      x1.x = p[16];
      x1.y = p[lda + 16];
    } else {
      x0 = *(const v2f*)p;                       // 8B-aligned b64 load
      x1 = *(const v2f*)(p + (size_t)16 * lda);
    }
  };
  auto loadB = [&](const float* p, v2f& x0, v2f& x1) {
    x0.x = p[0];
    x0.y = p[ldb];
    x1.x = p[16];
    x1.y = p[ldb + 16];
  };

  // ---- software pipeline: prologue ----
  v2f a0, a1, b0, b1;
  loadA(Ap, a0, a1);
  loadB(Bp, b0, b1);
  Ap += astep;
  Bp += bstep;

  // ---- steady state: prefetch k+4 while computing k ----
  for (int k = 4; k < K; k += 4) {
    v2f na0, na1, nb0, nb1;
    loadA(Ap, na0, na1);
    loadB(Bp, nb0, nb1);
    Ap += astep;
    Bp += bstep;

    acc00 = __builtin_amdgcn_wmma_f32_16x16x4_f32(false, a0, false, b0, (short)0, acc00, false, false);
    acc01 = __builtin_amdgcn_wmma_f32_16x16x4_f32(false, a0, false, b1, (short)0, acc01, false, false);
    acc10 = __builtin_amdgcn_wmma_f32_16x16x4_f32(false, a1, false, b0, (short)0, acc10, false, false);
    acc11 = __builtin_amdgcn_wmma_f32_16x16x4_f32(false, a1, false, b1, (short)0, acc11, false, false);

    a0 = na0; a1 = na1; b0 = nb0; b1 = nb1;
  }

  // ---- epilogue WMMAs ----
  acc00 = __builtin_amdgcn_wmma_f32_16x16x4_f32(false, a0, false, b0, (short)0, acc00, false, false);
  acc01 = __builtin_amdgcn_wmma_f32_16x16x4_f32(false, a0, false, b1, (short)0, acc01, false, false);
  acc10 = __builtin_amdgcn_wmma_f32_16x16x4_f32(false, a1, false, b0, (short)0, acc10, false, false);
  acc11 = __builtin_amdgcn_wmma_f32_16x16x4_f32(false, a1, false, b1, (short)0, acc11, false, false);

  // C/D layout: VGPR r -> row m0 + r + 8*(lane>=16), col n0 + (lane&15)
  const int rb = m0 + ((lane >> 4) << 3);
  const int cn0 = n0 + mr;
  const int cn1 = n0 + 16 + mr;
  float bi0 = 0.f, bi1 = 0.f;
  if (EPI == EPI_BIAS_RELU || EPI == EPI_BIAS_RELU_ADD) {
    bi0 = bias[cn0];
    bi1 = bias[cn1];
  }
#pragma unroll
  for (int mi = 0; mi < 2; ++mi) {
    const v8f aL = mi ? acc10 : acc00;
    const v8f aR = mi ? acc11 : acc01;
    const int rbase = rb + 16 * mi;
#pragma unroll
    for (int r = 0; r < 8; ++r) {
      const int row = rbase + r;
      float v0 = aL[r], v1 = aR[r];
      if (EPI == EPI_BIAS_RELU || EPI == EPI_BIAS_RELU_ADD) {
        v0 = fmaxf(v0 + bi0, 0.f);
        v1 = fmaxf(v1 + bi1, 0.f);
      }
      if (EPI == EPI_ADDX || EPI == EPI_BIAS_RELU_ADD) {
        v0 += X[(size_t)row * ldx + cn0];
        v1 += X[(size_t)row * ldx + cn1];
      }
      if (EPI == EPI_BOOL) {
        v0 = (v0 > 0.f) ? 1.f : 0.f;
        v1 = (v1 > 0.f) ? 1.f : 0.f;
      }
      C[(size_t)row * ldc + cn0] = v0;
      C[(size_t)row * ldc + cn1] = v1;
    }
  }
}

// ---------------- elementwise / small kernels ----------------

__global__ void binarize_kernel(const float* in, float* out, int n) {
  int e = blockIdx.x * blockDim.x + threadIdx.x;
  if (e < n) out[e] = (in[e] > 0.f) ? 1.f : 0.f;
}

__global__ void fzero_kernel(float* p, int n) {
  int e = blockIdx.x * blockDim.x + threadIdx.x;
  if (e < n) p[e] = 0.f;
}

__global__ void izero_kernel(int* p, int n) {
  int e = blockIdx.x * blockDim.x + threadIdx.x;
  if (e < n) p[e] = 0;
}

__global__ void add_kernel(const float* a, const float* b, float* c, int n) {
  int e = blockIdx.x * blockDim.x + threadIdx.x;
  if (e < n) c[e] = a[e] + b[e];
}

// dinv[i] = deg!=0 ? deg^-0.5 : 0 — one wave per row (wave32)
__global__ void row_dinv_kernel(const float* A, float* dinv, int n) {
  int row = blockIdx.x * (blockDim.x >> 5) + (threadIdx.x >> 5);
  int lane = threadIdx.x & 31;
  if (row >= n) return;
  float s = 0.f;
  for (int j = lane; j < n; j += 32) s += A[(size_t)row * n + j];
  for (int o = 16; o > 0; o >>= 1) s += __shfl_xor(s, o, 32);
  if (lane == 0) dinv[row] = (s != 0.f) ? (1.f / sqrtf(s)) : 0.f;
}

__global__ void lap_fill_kernel(const float* A, const float* dinv, float* Ln, int n) {
  int e = blockIdx.x * blockDim.x + threadIdx.x;
  if (e >= n * n) return;
  int i = e / n, j = e - i * n;
  float v = -dinv[i] * A[e] * dinv[j];
  if (i == j) v += 1.f;
  Ln[e] = v;
}

__global__ void sym_kernel(float* M, int n) {
  int e = blockIdx.x * blockDim.x + threadIdx.x;
  if (e >= n * n) return;
  int i = e / n, j = e - i * n;
  if (i < j) {
    float a = M[(size_t)i * n + j], b = M[(size_t)j * n + i];
    float v = 0.5f * (a + b);
    M[(size_t)i * n + j] = v;
    M[(size_t)j * n + i] = v;
  }
}

// next-level mask: Mo[r,c] = U[idx[r], idx[c]]  (U already boolean 0/1)
__global__ void gather_mask_kernel(const float* U, int ldu, const int* idx,
                                   float* Mo, int m) {
  int e = blockIdx.x * blockDim.x + threadIdx.x;
  if (e >= m * m) return;
  int r = e / m, c = e - r * m;
  Mo[e] = U[(size_t)idx[r] * ldu + idx[c]];
}

// scores = sigmoid(h @ w + b) — one wave per row
__global__ void pool_score_kernel(const float* h, const float* w, const float* b,
                                  float* sc, int n, int dim) {
  int row = blockIdx.x * (blockDim.x >> 5) + (threadIdx.x >> 5);
  int lane = threadIdx.x & 31;
  if (row >= n) return;
  float acc = 0.f;
  for (int k = lane; k < dim; k += 32) acc += h[(size_t)row * dim + k] * w[k];
  for (int o = 16; o > 0; o >>= 1) acc += __shfl_xor(acc, o, 32);
  if (lane == 0) sc[row] = 1.f / (1.f + expf(-(acc + b[0])));
}

__global__ void gather_scale_kernel(const float* hin, const int* idx,
                                    const float* sc, float* hout, int m, int dim) {
  int r = blockIdx.y;
  int f = blockIdx.x * blockDim.x + threadIdx.x;
  if (r >= m || f >= dim) return;
  int s = idx[r];
  hout[(size_t)r * dim + f] = hin[(size_t)s * dim + f] * sc[s];
}

__global__ void scatter_rows_kernel(const float* hin, const int* idx,
                                    float* hout, int m, int dim) {
  int r = blockIdx.y;
  int f = blockIdx.x * blockDim.x + threadIdx.x;
  if (r >= m || f >= dim) return;
  hout[(size_t)idx[r] * dim + f] = hin[(size_t)r * dim + f];
}

// ---------------- greedy eigh step: cyclic Jacobi on R = LK[rc,rc] ----------
// Single workgroup. Reproduces: eigh(R); sel = rc[argmax |V[0,:]|] with
// eigenvector columns in ascending-eigenvalue order (via rank of lambda).
#define JACOBI_SWEEPS 3

__global__ __launch_bounds__(256) void greedy_step_kernel(
    const float* __restrict__ LK, int ldk, int nfull,
    int* selected, int* idxOut, int step,
    float* R, float* Vv, int* rcBuf, int ld)
{
  __shared__ float shc, shs;
  __shared__ int sn;
  const int tid = threadIdx.x;

  if (tid == 0) {
    int c = 0;
    for (int i = 0; i < nfull; ++i)
      if (!selected[i]) rcBuf[c++] = i;
    sn = c;
  }
  __syncthreads();
  const int n = sn;  // == nfull - step

  for (int e = tid; e < n * n; e += blockDim.x) {
    int i = e / n, j = e - i * n;
    R[(size_t)i * ld + j] = LK[(size_t)rcBuf[i] * ldk + rcBuf[j]];
    Vv[(size_t)i * ld + j] = (i == j) ? 1.f : 0.f;
  }
  __syncthreads();

  for (int sweep = 0; sweep < JACOBI_SWEEPS; ++sweep) {
    for (int p = 0; p < n - 1; ++p) {
      for (int q = p + 1; q < n; ++q) {
        if (tid == 0) {
          float app = R[(size_t)p * ld + p];
          float aqq = R[(size_t)q * ld + q];
          float apq = R[(size_t)p * ld + q];
          float c = 1.f, s = 0.f;
          if (fabsf(apq) > 1e-12f) {
            float theta = 0.5f * (aqq - app) / apq;
            float t = (theta >= 0.f ? 1.f : -1.f) /
                      (fabsf(theta) + sqrtf(theta * theta + 1.f));
            c = 1.f / sqrtf(t * t + 1.f);
            s = t * c;
          }
          shc = c; shs = s;
        }
        __syncthreads();
        float c = shc, s = shs;
        if (s != 0.f) {  // block-uniform branch
          for (int i = tid; i < n; i += blockDim.x) {
            float rp = R[(size_t)i * ld + p], rq = R[(size_t)i * ld + q];
            R[(size_t)i * ld + p] = c * rp - s * rq;
            R[(size_t)i * ld + q] = s * rp + c * rq;
            float vp = Vv[(size_t)i * ld + p], vq = Vv[(size_t)i * ld + q];
            Vv[(size_t)i * ld + p] = c * vp - s * vq;
            Vv[(size_t)i * ld + q] = s * vp + c * vq;
          }
          __syncthreads();
          for (int j = tid; j < n; j += blockDim.x) {
            float rp = R[(size_t)p * ld + j], rq = R[(size_t)q * ld + j];
            R[(size_t)p * ld + j] = c * rp - s * rq;
            R[(size_t)q * ld + j] = s * rp + c * rq;
          }
        }
        __syncthreads();
      }
    }
  }

  if (tid == 0) {
    int jstar = 0; float best = -1.f;
    for (int j = 0; j < n; ++j) {
      float v = fabsf(Vv[j]);  // row 0 of V
      if (v > best) { best = v; jstar = j; }
    }
    float dj = R[(size_t)jstar * ld + jstar];
    int kk = 0;  // rank of lambda_{jstar} ascending (eigh ordering)
    for (int i = 0; i < n; ++i) {
      float di = R[(size_t)i * ld + i];
      if (di < dj || (di == dj && i < jstar)) kk++;
    }
    int sel = rcBuf[kk];
    selected[sel] = 1;
    idxOut[step] = sel;
  }
}

// ---------------------------------------------------------------------------

extern "C" void kernel_launch(void* const* d_in, const int* in_sizes, int n_in,
                              void* d_out, int out_size, void* d_ws, size_t ws_size,
                              hipStream_t stream) {
  (void)in_sizes; (void)n_in; (void)out_size; (void)ws_size;
  const int N0 = 384, N1 = 192, N2 = 96, DIM = 1024;

  const float* g = (const float*)d_in[0];
  const float* h = (const float*)d_in[1];
  // params leaves in jax pytree order: bottom{W1,W2,b1,b2}, down[0..1]{...},
  // pool[0..1]{b,w}, up[0..1]{...}
  const float* BO_W1 = (const float*)d_in[2];
  const float* BO_W2 = (const float*)d_in[3];
  const float* BO_b1 = (const float*)d_in[4];
  const float* BO_b2 = (const float*)d_in[5];
  const float* D0_W1 = (const float*)d_in[6];
  const float* D0_W2 = (const float*)d_in[7];
  const float* D0_b1 = (const float*)d_in[8];
  const float* D0_b2 = (const float*)d_in[9];
  const float* D1_W1 = (const float*)d_in[10];
  const float* D1_W2 = (const float*)d_in[11];
  const float* D1_b1 = (const float*)d_in[12];
  const float* D1_b2 = (const float*)d_in[13];
  const float* P0_b  = (const float*)d_in[14];
  const float* P0_w  = (const float*)d_in[15];
  const float* P1_b  = (const float*)d_in[16];
  const float* P1_w  = (const float*)d_in[17];
  const float* U0_W1 = (const float*)d_in[18];
  const float* U0_W2 = (const float*)d_in[19];
  const float* U0_b1 = (const float*)d_in[20];
  const float* U0_b2 = (const float*)d_in[21];
  const float* U1_W1 = (const float*)d_in[22];
  const float* U1_W2 = (const float*)d_in[23];
  const float* U1_b1 = (const float*)d_in[24];
  const float* U1_b2 = (const float*)d_in[25];

  // workspace bump allocator (256B aligned)
  uintptr_t wp = (uintptr_t)d_ws;
  auto falloc = [&](size_t elems) -> float* {
    float* p = (float*)wp;
    wp += ((elems * sizeof(float) + 255) & ~(size_t)255);
    return p;
  };
  auto ialloc = [&](size_t elems) -> int* {
    int* p = (int*)wp;
    wp += ((elems * sizeof(int) + 255) & ~(size_t)255);
    return p;
  };

  float* m0 = falloc((size_t)N0 * N0);
  float* m1 = falloc((size_t)N1 * N1);
  float* m2 = falloc((size_t)N2 * N2);
  float* Ln = falloc((size_t)N0 * N0);
  float* P2 = falloc((size_t)N0 * N0);
  float* P4 = falloc((size_t)N0 * N0);
  float* LK = falloc((size_t)N0 * N0);
  float* U  = falloc((size_t)N0 * N0);
  float* R  = falloc((size_t)N0 * N0);
  float* Vv = falloc((size_t)N0 * N0);
  float* dinv   = falloc(N0);
  float* scores = falloc(N0);
  float* z   = falloc((size_t)N0 * DIM);
  float* t   = falloc((size_t)N0 * DIM);
  float* h0  = falloc((size_t)N0 * DIM);
  float* h1  = falloc((size_t)N1 * DIM);
  float* hb  = falloc((size_t)N2 * DIM);
  float* hA  = falloc((size_t)N1 * DIM);
  float* hB  = falloc((size_t)N2 * DIM);
  float* hu  = falloc((size_t)N1 * DIM);
  float* hu2 = falloc((size_t)N0 * DIM);
  int* selected = ialloc(N0);
  int* rcbuf    = ialloc(N0);
  int* idx0     = ialloc(N1);
  int* idx1     = ialloc(N2);

  float* out0 = (float*)d_out;                      // hs[0]: 192x1024
  float* out1 = out0 + (size_t)N1 * DIM;            // hs[1]: 384x1024
  float* out2 = out1 + (size_t)N0 * DIM;            // hs[2]: 384x1024

  const dim3 blk(256);
  auto ceilb = [](int n, int b) { return (n + b - 1) / b; };

  // GEMM grids: grid.x = ceil(N/256) (8 waves x 32 cols), grid.y = M/32
  // ---------------- structure: level 0 ----------------
  binarize_kernel<<<ceilb(N0 * N0, 256), blk, 0, stream>>>(g, m0, N0 * N0);
  row_dinv_kernel<<<ceilb(N0, 8), blk, 0, stream>>>(m0, dinv, N0);
  lap_fill_kernel<<<ceilb(N0 * N0, 256), blk, 0, stream>>>(m0, dinv, Ln, N0);
  // LK = Ln^8 (three WMMA squarings), then symmetrize
  wmma_gemm_f32<EPI_NONE, false><<<dim3(2, 12), blk, 0, stream>>>(
      Ln, N0, Ln, N0, nullptr, nullptr, 0, P2, N0, N0, N0, N0);
  wmma_gemm_f32<EPI_NONE, false><<<dim3(2, 12), blk, 0, stream>>>(
      P2, N0, P2, N0, nullptr, nullptr, 0, P4, N0, N0, N0, N0);
  wmma_gemm_f32<EPI_NONE, false><<<dim3(2, 12), blk, 0, stream>>>(
      P4, N0, P4, N0, nullptr, nullptr, 0, LK, N0, N0, N0, N0);
  sym_kernel<<<ceilb(N0 * N0, 256), blk, 0, stream>>>(LK, N0);
  izero_kernel<<<ceilb(N0, 256), blk, 0, stream>>>(selected, N0);
  for (int s = 0; s < N1; ++s)
    greedy_step_kernel<<<1, blk, 0, stream>>>(LK, N0, N0, selected, idx0, s,
                                              R, Vv, rcbuf, N0);
  // 2-hop boolean coarsening: U = (m0@m0 > 0); m1 = U[idx0, idx0]
  wmma_gemm_f32<EPI_BOOL, false><<<dim3(2, 12), blk, 0, stream>>>(
      m0, N0, m0, N0, nullptr, nullptr, 0, U, N0, N0, N0, N0);
  gather_mask_kernel<<<ceilb(N1 * N1, 256), blk, 0, stream>>>(U, N0, idx0, m1, N1);

  // ---------------- structure: level 1 ----------------
  row_dinv_kernel<<<ceilb(N1, 8), blk, 0, stream>>>(m1, dinv, N1);
  lap_fill_kernel<<<ceilb(N1 * N1, 256), blk, 0, stream>>>(m1, dinv, Ln, N1);
  wmma_gemm_f32<EPI_NONE, false><<<dim3(1, 6), blk, 0, stream>>>(
      Ln, N1, Ln, N1, nullptr, nullptr, 0, P2, N1, N1, N1, N1);
  wmma_gemm_f32<EPI_NONE, false><<<dim3(1, 6), blk, 0, stream>>>(
      P2, N1, P2, N1, nullptr, nullptr, 0, P4, N1, N1, N1, N1);
  wmma_gemm_f32<EPI_NONE, false><<<dim3(1, 6), blk, 0, stream>>>(
      P4, N1, P4, N1, nullptr, nullptr, 0, LK, N1, N1, N1, N1);
  sym_kernel<<<ceilb(N1 * N1, 256), blk, 0, stream>>>(LK, N1);
  izero_kernel<<<ceilb(N1, 256), blk, 0, stream>>>(selected, N1);
  for (int s = 0; s < N2; ++s)
    greedy_step_kernel<<<1, blk, 0, stream>>>(LK, N1, N1, selected, idx1, s,
                                              R, Vv, rcbuf, N0);
  wmma_gemm_f32<EPI_BOOL, false><<<dim3(1, 6), blk, 0, stream>>>(
      m1, N1, m1, N1, nullptr, nullptr, 0, U, N1, N1, N1, N1);
  gather_mask_kernel<<<ceilb(N2 * N2, 256), blk, 0, stream>>>(U, N1, idx1, m2, N2);

  // ---------------- forward: down 0 ----------------
  wmma_gemm_f32<EPI_ADDX, true><<<dim3(4, 12), blk, 0, stream>>>(
      m0, N0, h, DIM, nullptr, h, DIM, z, DIM, N0, DIM, N0);
  wmma_gemm_f32<EPI_BIAS_RELU, false><<<dim3(4, 12), blk, 0, stream>>>(
      z, DIM, D0_W1, DIM, D0_b1, nullptr, 0, t, DIM, N0, DIM, DIM);
  wmma_gemm_f32<EPI_BIAS_RELU, false><<<dim3(4, 12), blk, 0, stream>>>(
      t, DIM, D0_W2, DIM, D0_b2, nullptr, 0, h0, DIM, N0, DIM, DIM);
  pool_score_kernel<<<ceilb(N0, 8), blk, 0, stream>>>(h0, P0_w, P0_b, scores, N0, DIM);
  gather_scale_kernel<<<dim3(4, N1), blk, 0, stream>>>(h0, idx0, scores, hA, N1, DIM);

  // ---------------- forward: down 1 ----------------
  wmma_gemm_f32<EPI_ADDX, true><<<dim3(4, 6), blk, 0, stream>>>(
      m1, N1, hA, DIM, nullptr, hA, DIM, z, DIM, N1, DIM, N1);
  wmma_gemm_f32<EPI_BIAS_RELU, false><<<dim3(4, 6), blk, 0, stream>>>(
      z, DIM, D1_W1, DIM, D1_b1, nullptr, 0, t, DIM, N1, DIM, DIM);
  wmma_gemm_f32<EPI_BIAS_RELU, false><<<dim3(4, 6), blk, 0, stream>>>(
      t, DIM, D1_W2, DIM, D1_b2, nullptr, 0, h1, DIM, N1, DIM, DIM);
  pool_score_kernel<<<ceilb(N1, 8), blk, 0, stream>>>(h1, P1_w, P1_b, scores, N1, DIM);
  gather_scale_kernel<<<dim3(4, N2), blk, 0, stream>>>(h1, idx1, scores, hB, N2, DIM);

  // ---------------- forward: bottom ----------------
  wmma_gemm_f32<EPI_ADDX, true><<<dim3(4, 3), blk, 0, stream>>>(
      m2, N2, hB, DIM, nullptr, hB, DIM, z, DIM, N2, DIM, N2);
  wmma_gemm_f32<EPI_BIAS_RELU, false><<<dim3(4, 3), blk, 0, stream>>>(
      z, DIM, BO_W1, DIM, BO_b1, nullptr, 0, t, DIM, N2, DIM, DIM);
  wmma_gemm_f32<EPI_BIAS_RELU, false><<<dim3(4, 3), blk, 0, stream>>>(
      t, DIM, BO_W2, DIM, BO_b2, nullptr, 0, hb, DIM, N2, DIM, DIM);

  // ---------------- forward: up 0 (level 1) → hs[0] ----------------
  fzero_kernel<<<ceilb(N1 * DIM, 256), blk, 0, stream>>>(hu, N1 * DIM);
  scatter_rows_kernel<<<dim3(4, N2), blk, 0, stream>>>(hb, idx1, hu, N2, DIM);
  wmma_gemm_f32<EPI_ADDX, true><<<dim3(4, 6), blk, 0, stream>>>(
      m1, N1, hu, DIM, nullptr, hu, DIM, z, DIM, N1, DIM, N1);
  wmma_gemm_f32<EPI_BIAS_RELU, false><<<dim3(4, 6), blk, 0, stream>>>(
      z, DIM, U0_W1, DIM, U0_b1, nullptr, 0, t, DIM, N1, DIM, DIM);
  wmma_gemm_f32<EPI_BIAS_RELU_ADD, false><<<dim3(4, 6), blk, 0, stream>>>(
      t, DIM, U0_W2, DIM, U0_b2, h1, DIM, out0, DIM, N1, DIM, DIM);

  // ---------------- forward: up 1 (level 0) → hs[1], hs[2] ----------------
  fzero_kernel<<<ceilb(N0 * DIM, 256), blk, 0, stream>>>(hu2, N0 * DIM);
  scatter_rows_kernel<<<dim3(4, N1), blk, 0, stream>>>(out0, idx0, hu2, N1, DIM);
  wmma_gemm_f32<EPI_ADDX, true><<<dim3(4, 12), blk, 0, stream>>>(
      m0, N0, hu2, DIM, nullptr, hu2, DIM, z, DIM, N0, DIM, N0);
  wmma_gemm_f32<EPI_BIAS_RELU, false><<<dim3(4, 12), blk, 0, stream>>>(
      z, DIM, U1_W1, DIM, U1_b1, nullptr, 0, t, DIM, N0, DIM, DIM);
  wmma_gemm_f32<EPI_BIAS_RELU_ADD, false><<<dim3(4, 12), blk, 0, stream>>>(
      t, DIM, U1_W2, DIM, U1_b2, h0, DIM, out1, DIM, N0, DIM, DIM);
  add_kernel<<<ceilb(N0 * DIM, 256), blk, 0, stream>>>(out1, h, out2, N0 * DIM);
}